// Cholesky_from_z_2173253452240
// MI455X (gfx1250) — compile-verified
//
#include <hip/hip_runtime.h>
#include <stdint.h>

// Problem constants (match reference)
#define N_DIM   256
#define BATCH   512
#define M_TRI   (N_DIM * (N_DIM - 1) / 2)   // 32640
#define WAVES_PER_BLOCK 8
#define CHUNKS  (N_DIM / 32)                // 8 chunks of 32 columns per row

// Per-chunk consumer: staggered ASYNCcnt wait (loads complete in order), then
// wave32 prefix-product scan + sqrt + masked NT store.
template<int C>
__device__ __forceinline__ void do_chunk(const float (&wbuf)[CHUNKS][32],
                                         int lane, int i,
                                         float* __restrict__ outrow,
                                         float& carry)
{
    // Loads 0..C are complete once no more than (CHUNKS-1-C) remain outstanding.
    asm volatile("s_wait_asynccnt %0" :: "n"(CHUNKS - 1 - C) : "memory");

    const int j = C * 32 + lane;                 // column handled by this lane
    float z = wbuf[C][lane];
    if (j >= i) z = 0.0f;                        // off-row lanes: neutral element
    const float t = 1.0f - z * z;                // t == 1 for off-row lanes

    // Inclusive prefix product across the wave (log2(32) = 5 steps).
    float p = t;
    #pragma unroll
    for (int d = 1; d < 32; d <<= 1) {
        const float o = __shfl_up(p, d, 32);
        if (lane >= d) p *= o;
    }
    // Exclusive prefix for this lane.
    float e = __shfl_up(p, 1, 32);
    if (lane == 0) e = 1.0f;

    const float q   = carry * e;                 // Π_{k<j} (1 - z²[i,k])
    const float tot = __shfl(p, 31, 32);         // whole-chunk product

    const float x2  = z * z * q;                 // matches reference: z²·(1 - rowsum)
    float res = copysignf(__builtin_sqrtf(x2), z);
    if (j == i)      res = 1.0f;                 // diagonal
    else if (j > i)  res = 0.0f;                 // upper triangle

    __builtin_nontemporal_store(res, &outrow[j]);

    carry *= tot;
}

// One wave32 per (batch, row). All 8 input-chunk DMAs (global -> LDS, ASYNCcnt)
// are issued up front for max memory-level parallelism, then consumed in order.
__global__ __launch_bounds__(256)
void cholz_scan_kernel(const float* __restrict__ vec, float* __restrict__ out)
{
    __shared__ float sbuf[WAVES_PER_BLOCK][CHUNKS][32];   // 8 KB/block

    const int lane = threadIdx.x & 31;
    const int wave = threadIdx.x >> 5;
    const int r    = blockIdx.x * WAVES_PER_BLOCK + wave;  // row id in [0, BATCH*N_DIM)
    const int b    = r >> 8;          // batch
    const int i    = r & (N_DIM - 1); // row index within matrix

    // Packed strictly-lower input: row i starts at i*(i-1)/2 (row-major tril order).
    const size_t inbase  = (size_t)b * M_TRI + (size_t)(i * (i - 1) / 2);
    const size_t outbase = (size_t)r * N_DIM;

    // LDS byte address of this lane's chunk-0 slot: low 32 bits of the flat
    // address of a __shared__ object are the workgroup-relative LDS byte offset.
    const uint32_t lds0 = (uint32_t)(uintptr_t)(&sbuf[wave][0][lane]);

    const int cap = (i > 0) ? (i - 1) : 0;   // clamp target, always a valid element

    // Issue ALL chunk DMAs back-to-back: 8 outstanding async loads per wave.
    #pragma unroll
    for (int c = 0; c < CHUNKS; ++c) {
        const int j  = c * 32 + lane;
        const int jj = (j < i) ? j : cap;                  // clamped in-bounds offset
        const float* gaddr    = vec + inbase + jj;
        const uint32_t laddr  = lds0 + (uint32_t)(c * 32 * 4);
        asm volatile("global_load_async_to_lds_b32 %0, %1, off"
                     :: "v"(laddr), "v"(gaddr)
                     : "memory");
    }

    float* outrow = out + outbase;
    float carry = 1.0f;   // running product of (1 - z²) over all prior columns

    do_chunk<0>(sbuf[wave], lane, i, outrow, carry);
    do_chunk<1>(sbuf[wave], lane, i, outrow, carry);
    do_chunk<2>(sbuf[wave], lane, i, outrow, carry);
    do_chunk<3>(sbuf[wave], lane, i, outrow, carry);
    do_chunk<4>(sbuf[wave], lane, i, outrow, carry);
    do_chunk<5>(sbuf[wave], lane, i, outrow, carry);
    do_chunk<6>(sbuf[wave], lane, i, outrow, carry);
    do_chunk<7>(sbuf[wave], lane, i, outrow, carry);
}

extern "C" void kernel_launch(void* const* d_in, const int* in_sizes, int n_in,
                              void* d_out, int out_size, void* d_ws, size_t ws_size,
                              hipStream_t stream)
{
    (void)in_sizes; (void)n_in; (void)out_size; (void)d_ws; (void)ws_size;
    const float* vec = (const float*)d_in[0];
    float* out = (float*)d_out;

    const int rows = BATCH * N_DIM;                    // 131072 rows -> 131072 waves
    dim3 grid(rows / WAVES_PER_BLOCK);                 // 16384 blocks
    dim3 block(WAVES_PER_BLOCK * 32);                  // 8 wave32s / block
    cholz_scan_kernel<<<grid, block, 0, stream>>>(vec, out);
}